// HeteroGATSkipLayer_85152021611241
// MI455X (gfx1250) — compile-verified
//
#include <hip/hip_runtime.h>
#include <math.h>

// ---------------------------------------------------------------------------
// HeteroGAT layer for MI455X (gfx1250, wave32).
//   GEMMs   : V_WMMA_F32_16X16X4_F32 (fp32 matrix path), 16x16 tile / wave.
//   Edge ops: VALU + global float atomics (L2-resident working set).
// ---------------------------------------------------------------------------

typedef __attribute__((ext_vector_type(2))) float v2f;
typedef __attribute__((ext_vector_type(8))) float v8f;

#define IN_FEATS 128
#define OUT_DIM  256     // 4 heads * 64
#define NHEADS   4
#define HDIM     64
#define NEG_SLOPE 0.2f

// ------------------------------ utility fills ------------------------------
__global__ void fill_kernel(float* __restrict__ p, float v, int n) {
    int i = blockIdx.x * blockDim.x + threadIdx.x;
    if (i < n) p[i] = v;
}

// Fold attention vector into projection:  out[k*4+h] = sum_d W[k, h*64+d] * a[h*64+d]
__global__ void fold_kernel(const float* __restrict__ W, const float* __restrict__ av,
                            float* __restrict__ out) {
    int tid = blockIdx.x * blockDim.x + threadIdx.x;
    if (tid >= IN_FEATS * NHEADS) return;
    int k = tid >> 2, h = tid & 3;
    float s = 0.f;
    #pragma unroll 8
    for (int d = 0; d < HDIM; ++d)
        s += W[(size_t)k * OUT_DIM + h * HDIM + d] * av[h * HDIM + d];
    out[tid] = s;
}

// el/er GEMV: out[i,h] = x[i,:] @ wf[:,h]   (wf is [128][4]); one wave per node
__global__ void gemv4_kernel(const float* __restrict__ x, const float* __restrict__ wf,
                             float* __restrict__ out, int N) {
    int wid  = (blockIdx.x * blockDim.x + threadIdx.x) >> 5;
    int lane = threadIdx.x & 31;
    if (wid >= N) return;
    const float4 xv = *(const float4*)(x + (size_t)wid * IN_FEATS + lane * 4);
    const float xs[4] = {xv.x, xv.y, xv.z, xv.w};
    float acc[4] = {0.f, 0.f, 0.f, 0.f};
    const float* w = wf + lane * 16;   // rows lane*4 .. lane*4+3
    #pragma unroll
    for (int j = 0; j < 4; ++j)
        #pragma unroll
        for (int h = 0; h < 4; ++h)
            acc[h] += xs[j] * w[j * 4 + h];
    #pragma unroll
    for (int off = 16; off > 0; off >>= 1)
        #pragma unroll
        for (int h = 0; h < 4; ++h)
            acc[h] += __shfl_xor(acc[h], off, 32);
    if (lane == 0)
        *(float4*)(out + (size_t)wid * 4) = make_float4(acc[0], acc[1], acc[2], acc[3]);
}

// ------------------------------ WMMA fp32 GEMM -----------------------------
// C[M x 256] = A[M x 128] @ W[128 x 256]  (+ bias0 + bias1 per column)
// One wave computes one 16x16 tile via 32 x V_WMMA_F32_16X16X4_F32.
// Tile layouts per CDNA5 ISA 7.12.2:
//   A (16x4 f32): lanes 0-15 hold K=0,1 ; lanes 16-31 hold K=2,3 (M = lane%16)
//   B (4x16 f32): symmetric, N = lane%16
//   C (16x16 f32): VGPR v, lane l -> row = 8*(l/16)+v, col = l%16
__global__ void gemm_wmma_f32(const float* __restrict__ A, const float* __restrict__ W,
                              const float* __restrict__ bias0, const float* __restrict__ bias1,
                              float* __restrict__ C, int M) {
    const int wave = threadIdx.x >> 5;
    const int lane = threadIdx.x & 31;
    const int t    = blockIdx.x * (blockDim.x >> 5) + wave;   // wave-uniform
    const int numTiles = (M >> 4) << 4;                       // == M when M%16==0
    if (t >= numTiles) return;                                // wave-uniform branch
    const int m0   = (t >> 4) << 4;
    const int n0   = (t & 15) << 4;
    const int half = lane >> 4;
    const int l16  = lane & 15;

    const float* arow = A + (size_t)(m0 + l16) * IN_FEATS;
    const float* wcol = W + n0 + l16;

    v8f c = {};
    #pragma unroll
    for (int k0 = 0; k0 < IN_FEATS; k0 += 4) {
        const int ka = k0 + 2 * half;
        v2f a, b;
        a.x = arow[ka];
        a.y = arow[ka + 1];
        b.x = wcol[(size_t)ka * OUT_DIM];
        b.y = wcol[(size_t)(ka + 1) * OUT_DIM];
        c = __builtin_amdgcn_wmma_f32_16x16x4_f32(false, a, false, b,
                                                  (short)0, c, false, false);
    }

    float bv = 0.f;
    if (bias0) bv += bias0[n0 + l16];
    if (bias1) bv += bias1[n0 + l16];
    float* crow = C + (size_t)(m0 + 8 * half) * OUT_DIM + n0 + l16;
    #pragma unroll
    for (int v = 0; v < 8; ++v)
        crow[(size_t)v * OUT_DIM] = c[v] + bv;
}

// ------------------------------ edge kernels -------------------------------
// logits: e = leakyrelu(el[src]+er[dst]); store; float atomic-max into m[dst]
__global__ void edge_logits_kernel(const int* __restrict__ src, const int* __restrict__ dst,
                                   const float* __restrict__ el, const float* __restrict__ er,
                                   float* __restrict__ elog, float* __restrict__ m, int E) {
    int tid = blockIdx.x * blockDim.x + threadIdx.x;
    if (tid >= E * NHEADS) return;
    int e = tid >> 2, h = tid & 3;
    int s = src[e], d = dst[e];
    float v = el[(size_t)s * 4 + h] + er[(size_t)d * 4 + h];
    v = v > 0.f ? v : NEG_SLOPE * v;
    elog[tid] = v;
    // float atomic max via sign-split (m initialized to -inf)
    float* addr = m + (size_t)d * 4 + h;
    if (v >= 0.f) atomicMax((int*)addr, __float_as_int(v));
    else          atomicMin((unsigned int*)addr, (unsigned int)__float_as_int(v));
}

// exp pass: ee = exp(e - m[dst]); z[dst] += ee
__global__ void edge_exp_kernel(const int* __restrict__ dst, float* __restrict__ elog,
                                const float* __restrict__ m, float* __restrict__ z, int E) {
    int tid = blockIdx.x * blockDim.x + threadIdx.x;
    if (tid >= E * NHEADS) return;
    int e = tid >> 2, h = tid & 3;
    int d = dst[e];
    float ex = __expf(elog[tid] - m[(size_t)d * 4 + h]);
    elog[tid] = ex;
    atomicAdd(z + (size_t)d * 4 + h, ex);
}

// aggregation: one wave per edge; lane handles 8 consecutive channels.
// out[dst, :] += (ee/z[dst]) * fs[src, :]   (256 floats / edge)
__global__ void aggregate_kernel(const int* __restrict__ src, const int* __restrict__ dst,
                                 const float* __restrict__ ee, const float* __restrict__ z,
                                 const float* __restrict__ fs, float* __restrict__ out, int E) {
    int wid  = (blockIdx.x * blockDim.x + threadIdx.x) >> 5;
    int lane = threadIdx.x & 31;
    if (wid >= E) return;
    int s = src[wid], d = dst[wid];
    int h = lane >> 3;   // 8 lanes per head
    float alpha = ee[(size_t)wid * 4 + h] / fmaxf(z[(size_t)d * 4 + h], 1e-12f);
    const float4* fp = (const float4*)(fs + (size_t)s * OUT_DIM + lane * 8);
    float4 f0 = fp[0], f1 = fp[1];
    float* op = out + (size_t)d * OUT_DIM + lane * 8;
    atomicAdd(op + 0, alpha * f0.x);
    atomicAdd(op + 1, alpha * f0.y);
    atomicAdd(op + 2, alpha * f0.z);
    atomicAdd(op + 3, alpha * f0.w);
    atomicAdd(op + 4, alpha * f1.x);
    atomicAdd(op + 5, alpha * f1.y);
    atomicAdd(op + 6, alpha * f1.z);
    atomicAdd(op + 7, alpha * f1.w);
}

// ------------------------------ host driver --------------------------------
extern "C" void kernel_launch(void* const* d_in, const int* in_sizes, int n_in,
                              void* d_out, int out_size, void* d_ws, size_t ws_size,
                              hipStream_t stream) {
    const float* x_flow  = (const float*)d_in[0];
    const float* x_ep    = (const float*)d_in[1];
    const int*   e1_src  = (const int*)d_in[2];
    const int*   e1_dst  = (const int*)d_in[3];
    const int*   e2_src  = (const int*)d_in[4];
    const int*   e2_dst  = (const int*)d_in[5];
    const float* W1      = (const float*)d_in[6];
    const float* al1     = (const float*)d_in[7];
    const float* ar1     = (const float*)d_in[8];
    const float* b1      = (const float*)d_in[9];
    const float* W2      = (const float*)d_in[10];
    const float* al2     = (const float*)d_in[11];
    const float* ar2     = (const float*)d_in[12];
    const float* b2      = (const float*)d_in[13];
    const float* Wr_flow = (const float*)d_in[14];
    const float* br_flow = (const float*)d_in[15];
    const float* Wr_ep   = (const float*)d_in[16];
    const float* br_ep   = (const float*)d_in[17];

    const int Nf = in_sizes[0] / IN_FEATS;   // 100000
    const int Ne = in_sizes[1] / IN_FEATS;   // 20000
    const int E1 = in_sizes[2];              // 1M
    const int E2 = in_sizes[4];              // 1M

    // ---- workspace carve-out (all offsets 64-float = 256B aligned) ----
    float* ws = (float*)d_ws;
    size_t off = 0;
    auto carve = [&](size_t n) { float* p = ws + off; off += (n + 63) & ~(size_t)63; return p; };
    float* fs1  = carve((size_t)Ne * OUT_DIM);
    float* fs2  = carve((size_t)Nf * OUT_DIM);
    float* el1  = carve((size_t)Ne * 4);
    float* er1  = carve((size_t)Nf * 4);
    float* el2  = carve((size_t)Nf * 4);
    float* er2  = carve((size_t)Ne * 4);
    float* m1   = carve((size_t)Nf * 4);
    float* z1   = carve((size_t)Nf * 4);
    float* m2   = carve((size_t)Ne * 4);
    float* z2   = carve((size_t)Ne * 4);
    float* ee1  = carve((size_t)E1 * 4);
    float* ee2  = carve((size_t)E2 * 4);
    float* Wel1 = carve(512);
    float* Wer1 = carve(512);
    float* Wel2 = carve(512);
    float* Wer2 = carve(512);

    float* h_flow = (float*)d_out;
    float* h_ep   = h_flow + (size_t)Nf * OUT_DIM;

    const float NEG_INF = -__builtin_huge_valf();

    // ---- init softmax stats ----
    fill_kernel<<<(Nf * 4 + 255) / 256, 256, 0, stream>>>(m1, NEG_INF, Nf * 4);
    fill_kernel<<<(Nf * 4 + 255) / 256, 256, 0, stream>>>(z1, 0.f,     Nf * 4);
    fill_kernel<<<(Ne * 4 + 255) / 256, 256, 0, stream>>>(m2, NEG_INF, Ne * 4);
    fill_kernel<<<(Ne * 4 + 255) / 256, 256, 0, stream>>>(z2, 0.f,     Ne * 4);

    // ---- fold attention vectors into 128x4 projections ----
    fold_kernel<<<2, 256, 0, stream>>>(W1, al1, Wel1);
    fold_kernel<<<2, 256, 0, stream>>>(W1, ar1, Wer1);
    fold_kernel<<<2, 256, 0, stream>>>(W2, al2, Wel2);
    fold_kernel<<<2, 256, 0, stream>>>(W2, ar2, Wer2);

    // ---- el / er GEMVs (wave per node) ----
    gemv4_kernel<<<(Ne + 7) / 8, 256, 0, stream>>>(x_ep,   Wel1, el1, Ne);
    gemv4_kernel<<<(Nf + 7) / 8, 256, 0, stream>>>(x_flow, Wer1, er1, Nf);
    gemv4_kernel<<<(Nf + 7) / 8, 256, 0, stream>>>(x_flow, Wel2, el2, Nf);
    gemv4_kernel<<<(Ne + 7) / 8, 256, 0, stream>>>(x_ep,   Wer2, er2, Ne);

    // ---- WMMA GEMMs: source features + residual init (bias folded in) ----
    // numTiles == M (M multiple of 16); 8 waves/block -> grid = ceil(M/8)
    gemm_wmma_f32<<<(Ne + 7) / 8, 256, 0, stream>>>(x_ep,   W1, nullptr, nullptr, fs1, Ne);
    gemm_wmma_f32<<<(Nf + 7) / 8, 256, 0, stream>>>(x_flow, W2, nullptr, nullptr, fs2, Nf);
    gemm_wmma_f32<<<(Nf + 7) / 8, 256, 0, stream>>>(x_flow, Wr_flow, br_flow, b1, h_flow, Nf);
    gemm_wmma_f32<<<(Ne + 7) / 8, 256, 0, stream>>>(x_ep,   Wr_ep,   br_ep,   b2, h_ep,   Ne);

    // ---- relation 1: endpoint -> flow ----
    edge_logits_kernel<<<(E1 * 4 + 255) / 256, 256, 0, stream>>>(e1_src, e1_dst, el1, er1, ee1, m1, E1);
    edge_exp_kernel<<<(E1 * 4 + 255) / 256, 256, 0, stream>>>(e1_dst, ee1, m1, z1, E1);
    aggregate_kernel<<<(E1 + 7) / 8, 256, 0, stream>>>(e1_src, e1_dst, ee1, z1, fs1, h_flow, E1);

    // ---- relation 2: flow -> endpoint ----
    edge_logits_kernel<<<(E2 * 4 + 255) / 256, 256, 0, stream>>>(e2_src, e2_dst, el2, er2, ee2, m2, E2);
    edge_exp_kernel<<<(E2 * 4 + 255) / 256, 256, 0, stream>>>(e2_dst, ee2, m2, z2, E2);
    aggregate_kernel<<<(E2 + 7) / 8, 256, 0, stream>>>(e2_src, e2_dst, ee2, z2, fs2, h_ep, E2);

    (void)n_in; (void)out_size; (void)ws_size;
}